// UnimodalBeta_82094004896340
// MI455X (gfx1250) — compile-verified
//
#include <hip/hip_runtime.h>
#include <hip/hip_bf16.h>
#include <math.h>

typedef __attribute__((ext_vector_type(2))) float v2f;
typedef __attribute__((ext_vector_type(8))) float v8f;

#define BATCH   131072
#define C_IN    512
#define NC      10
#define NGRID   550      // sum_{i=1..10} 10*i
#define EPSQ    1e-5f
#define LOG2E   1.44269504088896340736f

// ---------------------------------------------------------------------------
// Kernel 0: precompute (log2(t), log2(1-t)) for all 550 quadrature points.
// Segment i (1..10): n=10*i points, linspace(EPS, i/10-EPS, n).
// Cumulative points before segment s: 5*s*(s-1).
// ---------------------------------------------------------------------------
__global__ void build_tables_kernel(float2* __restrict__ tab) {
    int idx = blockIdx.x * blockDim.x + threadIdx.x;
    if (idx >= NGRID) return;
    int seg = 1;
    while (5 * seg * (seg + 1) <= idx) ++seg;      // seg covers [5s(s-1), 5s(s+1))
    int start = 5 * seg * (seg - 1);
    int j = idx - start;
    int n = 10 * seg;
    float b = (float)seg / 10.0f;
    float step = (b - 2.0f * EPSQ) / (float)(n - 1);
    float t = EPSQ + step * (float)j;
    float2 g;
    g.x = log2f(t);                 // log2(t)
    g.y = log1pf(-t) * LOG2E;       // log2(1-t)
    tab[idx] = g;
}

// ---------------------------------------------------------------------------
// Kernel 1: dual GEMV via V_WMMA_F32_16X16X4_F32.
// Block = 256 threads = 8 waves; each wave computes 16 rows of
// [alpha_raw, beta_raw] = x[16,512] @ [Wa | Wb | 0...] (16x16 C, cols 0/1 used)
// A fragment (16x4 f32): lane m in [0,16): row m, cols k..k+1;
//                        lane m+16:        row m, cols k+2..k+3.
// B fragment (4x16 f32): lane n in [0,16): col n, K rows k..k+1 in v[0],v[1];
//                        lane n+16:        col n, K rows k+2..k+3.
// ---------------------------------------------------------------------------
__global__ void __launch_bounds__(256) gemv_wmma_kernel(
    const float* __restrict__ x,
    const float* __restrict__ Wa,
    const float* __restrict__ Wb,
    float* __restrict__ alpha_raw,
    float* __restrict__ beta_raw) {
    __shared__ float sWa[C_IN];
    __shared__ float sWb[C_IN];

    int t = threadIdx.x;
    sWa[t]       = Wa[t];
    sWa[t + 256] = Wa[t + 256];
    sWb[t]       = Wb[t];
    sWb[t + 256] = Wb[t + 256];
    __syncthreads();

    const int wave = t >> 5;
    const int lane = t & 31;
    const int half = lane >> 4;     // 0: K pair (k,k+1), 1: K pair (k+2,k+3)
    const int n    = lane & 15;     // A: row-in-tile; B/C: column
    const long rowBase = ((long)blockIdx.x * 8 + wave) * 16;
    const float* __restrict__ xrow = x + (rowBase + n) * C_IN + 2 * half;

    v8f c = {0.f, 0.f, 0.f, 0.f, 0.f, 0.f, 0.f, 0.f};

    #pragma unroll 4
    for (int k = 0; k < C_IN; k += 4) {
        v2f a = *(const v2f*)(xrow + k);          // global_load_b64, 8B aligned
        int kk = k + 2 * half;
        v2f bfrag;
        bfrag.x = (n == 0) ? sWa[kk]     : ((n == 1) ? sWb[kk]     : 0.0f);
        bfrag.y = (n == 0) ? sWa[kk + 1] : ((n == 1) ? sWb[kk + 1] : 0.0f);
        c = __builtin_amdgcn_wmma_f32_16x16x4_f32(
                /*neg_a=*/false, a, /*neg_b=*/false, bfrag,
                /*c_mod=*/(short)0, c, /*reuse_a=*/false, /*reuse_b=*/false);
    }

    // C layout: lane n (0-15) VGPR j -> D[M=j][N=n]; lane n+16 -> D[M=j+8][N=n]
    if (n < 2) {
        float* __restrict__ dst = (n == 0) ? alpha_raw : beta_raw;
        long r0 = rowBase + 8 * half;
        #pragma unroll
        for (int j = 0; j < 8; ++j) dst[r0 + j] = c[j];
    }
}

// ---------------------------------------------------------------------------
// Kernel 2: per-row Beta integration + diff + normalize. 1 thread per row.
// probs_i ∝ S_i - S_{i-1}, S_i = sum over segment-i grid of
//           2^( (a-1)*log2 t + (b-1)*log2(1-t) - lbeta*log2e )
// (dx and any common scale cancel in the normalization; -lbeta kept for range)
// ---------------------------------------------------------------------------
__global__ void __launch_bounds__(256) beta_probs_kernel(
    const float* __restrict__ alpha_raw,
    const float* __restrict__ beta_raw,
    const float* __restrict__ b_alpha,
    const float* __restrict__ b_beta,
    const float2* __restrict__ tab,
    float* __restrict__ out) {
    __shared__ float2 stab[NGRID];
    for (int i = threadIdx.x; i < NGRID; i += 256) stab[i] = tab[i];
    __syncthreads();

    long row = (long)blockIdx.x * blockDim.x + threadIdx.x;
    float va = alpha_raw[row] + b_alpha[0];
    float vb = beta_raw[row]  + b_beta[0];
    // stable softplus: max(v,0) + log1p(exp(-|v|))
    float spa = fmaxf(va, 0.0f) + log1pf(__expf(-fabsf(va)));
    float spb = fmaxf(vb, 0.0f) + log1pf(__expf(-fabsf(vb)));
    float a = fminf(fmaxf(1.0f + spa, 1.0f), 100.0f);
    float b = fminf(fmaxf(1.0f + spb, 1.0f), 100.0f);

    float lbeta = lgammaf(a) + lgammaf(b) - lgammaf(a + b);
    float am1 = a - 1.0f, bm1 = b - 1.0f;
    float lb2 = lbeta * LOG2E;

    float p[NC];
    float prev = 0.0f;
    int idx = 0;
    #pragma unroll
    for (int seg = 1; seg <= NC; ++seg) {
        int nseg = 10 * seg;
        float s = 0.0f;
        for (int j = 0; j < nseg; ++j, ++idx) {
            float2 g = stab[idx];
            float e = fmaf(am1, g.x, fmaf(bm1, g.y, -lb2));
            s += exp2f(e);                      // v_exp_f32
        }
        p[seg - 1] = s - prev;
        prev = s;
    }
    float inv = 1.0f / prev;                    // telescoping sum == S_10
    #pragma unroll
    for (int i = 0; i < NC; ++i) out[row * NC + i] = p[i] * inv;
}

// ---------------------------------------------------------------------------
extern "C" void kernel_launch(void* const* d_in, const int* in_sizes, int n_in,
                              void* d_out, int out_size, void* d_ws, size_t ws_size,
                              hipStream_t stream) {
    const float* x  = (const float*)d_in[0];   // [B, 512]
    const float* Wa = (const float*)d_in[1];   // [512, 1]
    const float* ba = (const float*)d_in[2];   // [1]
    const float* Wb = (const float*)d_in[3];   // [512, 1]
    const float* bb = (const float*)d_in[4];   // [1]
    float* out = (float*)d_out;                // [B, 10]

    float*  ws        = (float*)d_ws;
    float*  alpha_raw = ws;                    // BATCH floats
    float*  beta_raw  = ws + BATCH;            // BATCH floats
    float2* tab       = (float2*)(ws + 2 * BATCH);  // 550 float2 (8B aligned)

    build_tables_kernel<<<(NGRID + 255) / 256, 256, 0, stream>>>(tab);
    gemv_wmma_kernel<<<BATCH / 128, 256, 0, stream>>>(x, Wa, Wb, alpha_raw, beta_raw);
    beta_probs_kernel<<<BATCH / 256, 256, 0, stream>>>(alpha_raw, beta_raw, ba, bb, tab, out);
}